// CLAPP_RSNN_74380243632769
// MI455X (gfx1250) — compile-verified
//
#include <hip/hip_runtime.h>
#include <hip/hip_bf16.h>

// ---------------------------------------------------------------------------
// CLAPP 2-layer SNN timestep for MI455X (gfx1250, wave32).
// B=32, NIN=NH=1024, beta=0.95, T=100.
// GEMMs use V_WMMA_F32_16X16X4_F32 (exact fp32); dW outer products use
// non-temporal b128 stores (256 MB write-once stream -> keep out of L2).
// Surrogate gradient uses hardware v_rcp_f32 instead of IEEE divide.
// ---------------------------------------------------------------------------

typedef float v2f __attribute__((ext_vector_type(2)));
typedef float v4f __attribute__((ext_vector_type(4)));
typedef float v8f __attribute__((ext_vector_type(8)));

#define BETA_F 0.95f
#define PI_F   3.14159265358979323846f
#define NROW   1024   // NIN == NH
#define BATCH  32

// workspace layout (float offsets)
#define WS_MU0   0
#define WS_MU1   32
#define WS_SPK0  64
#define WS_S0    (WS_SPK0 + BATCH * NROW)   // 32832
#define WS_T0    (WS_S0   + BATCH * NROW)   // 65600
#define WS_S1    (WS_T0   + BATCH * NROW)   // 98368
#define WS_T1    (WS_S1   + BATCH * NROW)   // 131136

// output layout (float offsets)
#define OUT_SPK1 0
#define OUT_MEM0 (BATCH * NROW)             // 32768
#define OUT_MEM1 (2 * BATCH * NROW)         // 65536
#define OUT_LOSS (3 * BATCH * NROW)         // 98304
#define OUT_DW0  (OUT_LOSS + 2)             // 98306
#define OUT_DW1  (OUT_DW0 + BATCH * NROW * NROW)

// ---------------------------------------------------------------------------
// Kernel 0: row means of prev_spk0/1 (blocks 0..63) and t0 = beta*trace0 + inp
// (blocks 64..191).
// ---------------------------------------------------------------------------
__global__ __launch_bounds__(256) void prep_kernel(
    const float* __restrict__ prev0, const float* __restrict__ prev1,
    const float* __restrict__ trace0, const float* __restrict__ inp,
    float* __restrict__ ws)
{
    __shared__ float red[256];
    const int blk = blockIdx.x;
    const int tid = threadIdx.x;
    if (blk < 64) {
        const float* row = (blk < 32) ? (prev0 + (size_t)blk * NROW)
                                      : (prev1 + (size_t)(blk - 32) * NROW);
        float s = 0.0f;
        #pragma unroll
        for (int i = tid; i < NROW; i += 256) s += row[i];
        red[tid] = s;
        __syncthreads();
        for (int off = 128; off > 0; off >>= 1) {
            if (tid < off) red[tid] += red[tid + off];
            __syncthreads();
        }
        if (tid == 0) ws[WS_MU0 + blk] = red[0] * (1.0f / (float)NROW);
    } else {
        const int idx = (blk - 64) * 256 + tid;   // 128 blocks * 256 = 32768
        ws[WS_T0 + idx] = BETA_F * trace0[idx] + inp[idx];
    }
}

// ---------------------------------------------------------------------------
// GEMM + LIF update. One wave computes a 16(batch) x 16(neuron) C tile:
//   C = X(32x1024) @ W^T,  via 256 x V_WMMA_F32_16X16X4_F32.
// A frag (16x4 f32, 2 VGPR): lanes 0-15 hold {K0,K1} of row M=lane,
//   lanes 16-31 hold {K2,K3} of row M=lane-16  -> one float2 load per lane.
// B frag (4x16 f32, 2 VGPR): B[k,n] = W[n,k], so lane holds the contiguous
//   pair W[n0+lane&15, k+2*(lane>>4) .. +1]   -> one float2 load per lane.
// Epilogue (C layout: vgpr r -> M = r + 8*(lane>>4), N = n0 + (lane&15)):
//   mem' = beta*mem + cur; spk = mem'>1; mem'' = mem'-spk
//   s = prev_spk * surrogate(mem''-1);  t_next = beta*trace_next + spk (L0 only)
// ---------------------------------------------------------------------------
__global__ __launch_bounds__(128) void snn_layer_kernel(
    const float* __restrict__ X, const float* __restrict__ W,
    const float* __restrict__ mem_in, const float* __restrict__ prev_spk,
    const float* __restrict__ trace_next,
    float* __restrict__ mem_out, float* __restrict__ spk_out,
    float* __restrict__ s_out, float* __restrict__ t_next_out)
{
    const int lane = threadIdx.x & 31;
    const int wg   = blockIdx.x * 4 + (threadIdx.x >> 5); // 128 waves total
    const int m0   = (wg >> 6) << 4;   // 0 or 16
    const int n0   = (wg & 63) << 4;   // 0..1008
    const int half = lane >> 4;        // 0 or 1
    const int lidx = lane & 15;

    const float* Arow = X + (size_t)(m0 + lidx) * NROW + 2 * half;
    const float* Brow = W + (size_t)(n0 + lidx) * NROW + 2 * half;

    v8f acc = {};
    #pragma unroll 8
    for (int k = 0; k < NROW; k += 4) {
        v2f a = *(const v2f*)(Arow + k);
        v2f b = *(const v2f*)(Brow + k);
        acc = __builtin_amdgcn_wmma_f32_16x16x4_f32(
            /*neg_a=*/false, a, /*neg_b=*/false, b,
            /*c_mod=*/(short)0, acc, /*reuse_a=*/false, /*reuse_b=*/false);
    }

    const int n = n0 + lidx;
    #pragma unroll
    for (int r = 0; r < 8; ++r) {
        const int m   = m0 + r + 8 * half;
        const int idx = m * NROW + n;
        float mv  = BETA_F * mem_in[idx] + acc[r];
        float spk = (mv > 1.0f) ? 1.0f : 0.0f;
        mv -= spk;
        mem_out[idx] = mv;
        spk_out[idx] = spk;
        const float xr = mv - 1.0f;
        const float px = PI_F * xr;
        // surrogate = 1 / (pi * (1 + px^2)); hardware v_rcp_f32 (~1 ulp)
        const float denom = PI_F * __builtin_fmaf(px, px, 1.0f);
        s_out[idx] = prev_spk[idx] * __builtin_amdgcn_rcpf(denom);
        if (t_next_out) t_next_out[idx] = BETA_F * trace_next[idx] + spk;
    }
}

// ---------------------------------------------------------------------------
// Loss: losses[l] = mean_b( -bf * sum_n spk[b,n]*(prev[b,n]-mu[b]) )
// Deterministic tree reduction, one block per layer.
// ---------------------------------------------------------------------------
__global__ __launch_bounds__(256) void loss_kernel(
    const float* __restrict__ ws, const float* __restrict__ spk1,
    const float* __restrict__ prev0, const float* __restrict__ prev1,
    const int* __restrict__ bfp, float* __restrict__ losses)
{
    __shared__ float red[256];
    const int l   = blockIdx.x;
    const int tid = threadIdx.x;
    const float* spk  = l ? spk1  : (ws + WS_SPK0);
    const float* prev = l ? prev1 : prev0;
    const float* mu   = ws + (l ? WS_MU1 : WS_MU0);

    float s = 0.0f;
    for (int idx = tid; idx < BATCH * NROW; idx += 256) {
        const int b = idx >> 10;
        s += spk[idx] * (prev[idx] - mu[b]);
    }
    red[tid] = s;
    __syncthreads();
    for (int off = 128; off > 0; off >>= 1) {
        if (tid < off) red[tid] += red[tid + off];
        __syncthreads();
    }
    if (tid == 0) losses[l] = (-(float)(*bfp) / (float)BATCH) * red[0];
}

// ---------------------------------------------------------------------------
// dW[b,i,j] = bf * s[b,i] * t[b,j].  One block per (layer,b,i) row; each lane
// writes one float4 with a non-temporal store (write-once stream, bypass L2
// retention).  65536 blocks x 256 threads -> 256 MB total.
// ---------------------------------------------------------------------------
__global__ __launch_bounds__(256) void dw_kernel(
    const float* __restrict__ ws, const int* __restrict__ bfp,
    float* __restrict__ dW0, float* __restrict__ dW1)
{
    const unsigned bid   = blockIdx.x;
    const int      layer = (int)(bid >> 15);
    const unsigned rem   = bid & 32767u;       // b*1024 + i
    const unsigned b     = rem >> 10;

    const float* s = ws + (layer ? WS_S1 : WS_S0);
    const float* t = ws + (layer ? WS_T1 : WS_T0);
    const float  sv = s[rem] * (float)(*bfp);

    float* dst = (layer ? dW1 : dW0) + ((size_t)rem << 10);
    const float* trow = t + ((size_t)b << 10);

    const int j = threadIdx.x << 2;
    v4f t4 = *(const v4f*)(trow + j);
    v4f o  = t4 * sv;
    __builtin_nontemporal_store(o, (v4f*)(dst + j));
}

// ---------------------------------------------------------------------------
extern "C" void kernel_launch(void* const* d_in, const int* in_sizes, int n_in,
                              void* d_out, int out_size, void* d_ws, size_t ws_size,
                              hipStream_t stream) {
    const float* inp   = (const float*)d_in[0];
    const float* W0    = (const float*)d_in[1];
    const float* W1    = (const float*)d_in[2];
    const float* mem0  = (const float*)d_in[3];
    const float* mem1  = (const float*)d_in[4];
    const float* prev0 = (const float*)d_in[5];
    const float* prev1 = (const float*)d_in[6];
    const float* tr0   = (const float*)d_in[7];
    const float* tr1   = (const float*)d_in[8];
    const int*   bfp   = (const int*)d_in[9];

    float* out = (float*)d_out;
    float* ws  = (float*)d_ws;

    float* o_spk1 = out + OUT_SPK1;
    float* o_mem0 = out + OUT_MEM0;
    float* o_mem1 = out + OUT_MEM1;
    float* o_loss = out + OUT_LOSS;
    float* o_dW0  = out + OUT_DW0;
    float* o_dW1  = out + OUT_DW1;

    // 1) row means of prev_spk + t0 = beta*trace0 + inp
    prep_kernel<<<192, 256, 0, stream>>>(prev0, prev1, tr0, inp, ws);

    // 2) layer 0: cur0 = inp @ W0^T (WMMA f32), LIF update, s0, t1
    snn_layer_kernel<<<32, 128, 0, stream>>>(
        inp, W0, mem0, prev0, tr1,
        o_mem0, ws + WS_SPK0, ws + WS_S0, ws + WS_T1);

    // 3) layer 1: cur1 = spk0 @ W1^T (WMMA f32), LIF update, s1
    snn_layer_kernel<<<32, 128, 0, stream>>>(
        ws + WS_SPK0, W1, mem1, prev1, nullptr,
        o_mem1, o_spk1, ws + WS_S1, nullptr);

    // 4) losses (deterministic reductions)
    loss_kernel<<<2, 256, 0, stream>>>(ws, o_spk1, prev0, prev1, bfp, o_loss);

    // 5) dW0/dW1 outer products, streaming stores (the 256 MB bottleneck)
    dw_kernel<<<65536, 256, 0, stream>>>(ws, bfp, o_dW0, o_dW1);
}